// QuantumLayer_25005299597863
// MI455X (gfx1250) — compile-verified
//
#include <hip/hip_runtime.h>

#define NQ 10
#define NL 4
#define WAVES_PER_BLOCK 8

typedef float v2 __attribute__((ext_vector_type(2)));

__device__ __forceinline__ float shx(float v, int mask) {
  return __shfl_xor(v, mask, 32);   // lane XOR permute
}
__device__ __forceinline__ float bperm(float v, int srcLane) {
  return __shfl(v, srcLane, 32);    // generic lane gather (ds_bpermute)
}

// ---- forced VOP3P packed-f32 math (accumulator tied: FMAC form) ----------
__device__ __forceinline__ v2 pk_mul(v2 a, v2 b) {
  v2 d;
  asm("v_pk_mul_f32 %0, %1, %2" : "=v"(d) : "v"(a), "v"(b));
  return d;
}
__device__ __forceinline__ v2 pk_fma(v2 a, v2 b, v2 c) {
  asm("v_pk_fma_f32 %0, %1, %2, %0" : "+v"(c) : "v"(a), "v"(b));
  return c;
}
// src1 half-swapped: c.lo += a.lo*b.hi ; c.hi += a.hi*b.lo
__device__ __forceinline__ v2 pk_fma_swapb(v2 a, v2 b, v2 c) {
  asm("v_pk_fma_f32 %0, %1, %2, %0 op_sel:[0,1,0] op_sel_hi:[1,0,1]"
      : "+v"(c) : "v"(a), "v"(b));
  return c;
}
// complex MAC: acc + u*a, with urr={ur,ur}, uii={-ui,+ui} (memory-baked)
__device__ __forceinline__ v2 cmac(v2 urr, v2 uii, v2 a, v2 acc) {
  return pk_fma_swapb(uii, a, pk_fma(urr, a, acc));
}
__device__ __forceinline__ v2 cmul(v2 urr, v2 uii, v2 a) {
  return pk_fma_swapb(uii, a, pk_mul(urr, a));
}

// Compile-time register-index map for the 4 register-register CNOTs
// (ctrl r4->r3, r3->r2, r2->r1, r1->r0 in program order; gather form:
// last program op innermost).
__host__ __device__ constexpr int rr_map(int j) {
  int v = j ^ ((j >> 1) & 1);        // CNOT r1->r0
  v ^= ((v >> 2) & 1) << 1;          // CNOT r2->r1
  v ^= ((v >> 3) & 1) << 2;          // CNOT r3->r2
  v ^= ((v >> 4) & 1) << 3;          // CNOT r4->r3
  return v;
}

// ---------------------------------------------------------------------------
// Gate record in d_ws: per gate 8 v2 values
//   {u00rr,u00ii, u01rr,u01ii, u10rr,u10ii, u11rr,u11ii}
// with urr = {re,re}, uii = {-im,+im}.
// ---------------------------------------------------------------------------

// 1-qubit gate on a "register bit" R (state bit 5+R): pairs live in one lane.
template<int R>
__device__ __forceinline__ void gate_reg(v2 (&st)[32], const v2* __restrict__ g) {
  const v2 u00rr = g[0], u00ii = g[1], u01rr = g[2], u01ii = g[3];
  const v2 u10rr = g[4], u10ii = g[5], u11rr = g[6], u11ii = g[7];
#pragma unroll
  for (int j = 0; j < 32; ++j) {
    if ((j >> R) & 1) continue;
    const int j1 = j | (1 << R);
    const v2 a0 = st[j], a1 = st[j1];
    st[j]  = cmac(u01rr, u01ii, a1, cmul(u00rr, u00ii, a0));
    st[j1] = cmac(u11rr, u11ii, a1, cmul(u10rr, u10ii, a0));
  }
}

// 1-qubit gate on a "lane bit" P (state bit P<5). Select-free:
//   new = uS*self + uP*partner,  uS = hi?u11:u00, uP = hi?u10:u01.
template<int P>
__device__ __forceinline__ void gate_lane(v2 (&st)[32], int lane,
                                          const v2* __restrict__ g) {
  const bool hi = (lane >> P) & 1;
  const v2 uSrr = hi ? g[6] : g[0];
  const v2 uSii = hi ? g[7] : g[1];
  const v2 uPrr = hi ? g[4] : g[2];
  const v2 uPii = hi ? g[5] : g[3];
#pragma unroll
  for (int j = 0; j < 32; ++j) {
    const v2 s = st[j];
    v2 p;
    p.x = shx(s.x, 1 << P);
    p.y = shx(s.y, 1 << P);
    st[j] = cmac(uPrr, uPii, p, cmul(uSrr, uSii, s));
  }
}

// Fused CNOT block: CNOT(0,1)..CNOT(8,9) in one gather.
//   dest register j <- source register rr_map(j) (compile time)
//   dest lane d     <- source lane i0 (even j) / i1 = i0^16 (odd j: folded
//                      reg-ctrl/lane-tgt CNOT(4,5))
__device__ __forceinline__ void cnot_block(v2 (&st)[32], int i0, int i1) {
  v2 nst[32];
#pragma unroll
  for (int j = 0; j < 32; ++j) {
    const int src = rr_map(j);
    const int sl  = (j & 1) ? i1 : i0;
    nst[j].x = bperm(st[src].x, sl);
    nst[j].y = bperm(st[src].y, sl);
  }
#pragma unroll
  for (int j = 0; j < 32; ++j) st[j] = nst[j];
}

// Ring-closing CNOT(9,0): lane-bit-0 control, register-bit-4 target.
__device__ __forceinline__ void cnot_ring(v2 (&st)[32], int lane) {
  const bool ctl = lane & 1;
#pragma unroll
  for (int j = 0; j < 16; ++j) {
    const int j1 = j | 16;
    const v2 t = st[j];
    st[j]  = ctl ? st[j1] : st[j];
    st[j1] = ctl ? t : st[j1];
  }
}

// ---------------------------------------------------------------------------
// Precompute 40 shared Rot matrices in pk-ready layout (16 floats per gate).
// ---------------------------------------------------------------------------
__global__ void qsim_prep_gates(const float* __restrict__ w, float* __restrict__ g) {
  const int idx = threadIdx.x;           // idx = l*10 + q
  if (idx >= NL * NQ) return;
  const float phi   = w[idx * 3 + 0];
  const float theta = w[idx * 3 + 1];
  const float omega = w[idx * 3 + 2];
  float c, s;
  sincosf(0.5f * theta, &s, &c);
  float sa, ca, sm, cm;
  sincosf(0.5f * (phi + omega), &sa, &ca);
  sincosf(0.5f * (phi - omega), &sm, &cm);
  const float u00r =  ca * c, u00i = -sa * c;   // e^{-i(phi+omega)/2} cos
  const float u01r = -cm * s, u01i = -sm * s;   // -e^{ i(phi-omega)/2} sin
  const float u10r =  cm * s, u10i = -sm * s;   // e^{-i(phi-omega)/2} sin
  const float u11r =  ca * c, u11i =  sa * c;   // e^{ i(phi+omega)/2} cos
  float* o = g + idx * 16;
  o[0]  = u00r; o[1]  = u00r; o[2]  = -u00i; o[3]  = u00i;
  o[4]  = u01r; o[5]  = u01r; o[6]  = -u01i; o[7]  = u01i;
  o[8]  = u10r; o[9]  = u10r; o[10] = -u10i; o[11] = u10i;
  o[12] = u11r; o[13] = u11r; o[14] = -u11i; o[15] = u11i;
}

// ---------------------------------------------------------------------------
// One wave32 per batch sample; 1024-amp complex state in VGPRs as {re,im}
// pairs. State index s: bits[4:0]=lane, bits[9:5]=register j. Qubit q <-> bit
// 9-q (wire 0 = MSB, as in the reference).
// ---------------------------------------------------------------------------
__global__ __launch_bounds__(WAVES_PER_BLOCK * 32)
void qsim_kernel(const float* __restrict__ x, const float* __restrict__ gates,
                 float* __restrict__ out, int batch) {
  const int lane = threadIdx.x & 31;
  const int wave = threadIdx.x >> 5;
  const int b = blockIdx.x * WAVES_PER_BLOCK + wave;
  if (b >= batch) return;

  // ---- Angle encoding: product state prod_q (bit ? sin(x/2) : cos(x/2)) ----
  float cq[5], sq[5];
  float f_lane = 1.0f;
#pragma unroll
  for (int q = 0; q < NQ; ++q) {
    const float a = 0.5f * x[b * NQ + q];
    float s, c;
    sincosf(a, &s, &c);
    if (q < 5) { cq[q] = c; sq[q] = s; }
    else       f_lane *= ((lane >> (9 - q)) & 1) ? s : c;
  }

  v2 st[32];
#pragma unroll
  for (int j = 0; j < 32; ++j) {
    float amp = f_lane;
#pragma unroll
    for (int q = 0; q < 5; ++q)
      amp *= ((j >> (4 - q)) & 1) ? sq[q] : cq[q];
    st[j].x = amp;
    st[j].y = 0.0f;
  }

  // Layer-invariant source-lane indices for the fused CNOT gather
  // (lane chain CNOT(5,6)..CNOT(8,9), last op innermost).
  int u = lane ^ ((lane >> 1) & 1);  // CNOT lane-bit1 -> bit0   (q=8)
  u ^= ((u >> 2) & 1) << 1;          // CNOT lane-bit2 -> bit1   (q=7)
  u ^= ((u >> 3) & 1) << 2;          // CNOT lane-bit3 -> bit2   (q=6)
  u ^= ((u >> 4) & 1) << 3;          // CNOT lane-bit4 -> bit3   (q=5)
  const int i0 = u;                  // even registers
  const int i1 = u ^ 16;             // odd registers: folded CNOT(4,5)

  // ---- 4 layers: 10 Rot gates + fused CNOT ring ----
  for (int l = 0; l < NL; ++l) {
    const v2* g = (const v2*)(gates + l * (NQ * 16));
    gate_reg<4>(st, g + 0 * 8);          // q=0 (state bit 9)
    gate_reg<3>(st, g + 1 * 8);
    gate_reg<2>(st, g + 2 * 8);
    gate_reg<1>(st, g + 3 * 8);
    gate_reg<0>(st, g + 4 * 8);
    gate_lane<4>(st, lane, g + 5 * 8);   // q=5 (state bit 4)
    gate_lane<3>(st, lane, g + 6 * 8);
    gate_lane<2>(st, lane, g + 7 * 8);
    gate_lane<1>(st, lane, g + 8 * 8);
    gate_lane<0>(st, lane, g + 9 * 8);
    cnot_block(st, i0, i1);              // CNOT(0,1)..CNOT(8,9) fused
    cnot_ring(st, lane);                 // CNOT(9,0)
  }

  // ---- Readout: <Z_q> = sum_s |amp|^2 * (1 - 2*bit_q(s)) ----
  float acc[NQ];
#pragma unroll
  for (int q = 0; q < NQ; ++q) acc[q] = 0.0f;
  float lsign[5];
#pragma unroll
  for (int p = 0; p < 5; ++p) lsign[p] = ((lane >> p) & 1) ? -1.0f : 1.0f;

#pragma unroll
  for (int j = 0; j < 32; ++j) {
    const float p2 = st[j].x * st[j].x + st[j].y * st[j].y;
#pragma unroll
    for (int q = 0; q < 5; ++q)
      acc[q] += ((j >> (4 - q)) & 1) ? -p2 : p2;
#pragma unroll
    for (int q = 5; q < NQ; ++q)
      acc[q] = fmaf(p2, lsign[9 - q], acc[q]);
  }

#pragma unroll
  for (int off = 16; off > 0; off >>= 1) {
#pragma unroll
    for (int q = 0; q < NQ; ++q) acc[q] += shx(acc[q], off);
  }

  if (lane == 0) {
#pragma unroll
    for (int q = 0; q < NQ; ++q) out[b * NQ + q] = acc[q];
  }
}

// ---------------------------------------------------------------------------
extern "C" void kernel_launch(void* const* d_in, const int* in_sizes, int n_in,
                              void* d_out, int out_size, void* d_ws, size_t ws_size,
                              hipStream_t stream) {
  const float* x = (const float*)d_in[0];       // [B, 10] f32
  const float* w = (const float*)d_in[1];       // [4, 10, 3] f32
  float* out = (float*)d_out;                   // [B, 10] f32
  float* gates = (float*)d_ws;                  // 40 * 16 f32 = 2560 B

  const int batch = in_sizes[0] / NQ;

  qsim_prep_gates<<<1, 64, 0, stream>>>(w, gates);

  const int blocks = (batch + WAVES_PER_BLOCK - 1) / WAVES_PER_BLOCK;
  qsim_kernel<<<blocks, WAVES_PER_BLOCK * 32, 0, stream>>>(x, gates, out, batch);
}